// encoder_5050881540748
// MI455X (gfx1250) — compile-verified
//
#include <hip/hip_runtime.h>
#include <hip/hip_bf16.h>

// ---------------------------------------------------------------------------
// Problem constants (match reference)
// ---------------------------------------------------------------------------
#define NN_  8192
#define DD_  512
#define HEADS_ 8
#define DHEAD_ 64
#define MM_  256          // landmarks
#define LL_  32           // rows averaged per landmark (N/M)
#define WP_  256
#define KCONV_ 33
#define QKVLD_ 1536       // 3*H*Dh row stride of qkv

typedef __attribute__((ext_vector_type(2))) float v2f;
typedef __attribute__((ext_vector_type(8))) float v8f;

#define WMMA_F32(a, b, c) \
  __builtin_amdgcn_wmma_f32_16x16x4_f32(false, (a), false, (b), (short)0, (c), false, false)

// store one 16x16 accumulator tile
template <int ACC>
__device__ __forceinline__ void store_tile16(float* __restrict__ C, int ldc,
                                             int m0, int n0, int lid, int half,
                                             const v8f& acc, float alpha,
                                             float bv, float beta) {
#pragma unroll
  for (int i = 0; i < 8; ++i) {
    const int m = m0 + i + half * 8;
    const size_t idx = (size_t)m * ldc + n0 + lid;
    float v = alpha * acc[i] + bv;
    if (ACC) v += beta * C[idx];
    C[idx] = v;
  }
}

// ---------------------------------------------------------------------------
// Generic f32 WMMA GEMM: one wave computes a 32x32 tile of
//   C = alpha * A @ op(B)  [+ bias[n]]  [+ beta * C_in]
// 2x2 register blocking over V_WMMA_F32_16X16X4_F32: 4 WMMAs per 2 A-frag +
// 2 B-frag loads (2x the L2 arithmetic intensity of 1x1 tiling), and 4
// independent accumulator chains so consecutive WMMAs have no RAW hazard.
// A: row-major M x K (lda). TRANS_B: B given as [N,K] row-major (B^T),
// else B is [K,N] row-major. M,N multiples of 32; K multiple of 32.
// Fragment layouts per CDNA5 ISA:
//   A 16x4 : lane m = l&15, K-pair = (l>>4)*2  (VGPR0=K0/K2, VGPR1=K1/K3)
//   B 4x16 : lane n = l&15, same K-pair
//   C 16x16: VGPR i holds M = i + 8*(l>>4), N = l&15
// ---------------------------------------------------------------------------
template <int TRANS_B, int ADD_BIAS, int ACCUM>
__global__ __launch_bounds__(32)
void wmma_gemm_f32(const float* __restrict__ A, int lda,
                   const float* __restrict__ B, int ldb,
                   float* __restrict__ C, int ldc,
                   const float* __restrict__ bias,
                   int K, float alpha, float beta) {
  const int lane = threadIdx.x & 31;
  const int lid  = lane & 15;
  const int half = lane >> 4;
  const int tm = blockIdx.y * 32;
  const int tn = blockIdx.x * 32;

  const float* a0p = A + (size_t)(tm + lid) * lda + half * 2;
  const float* a1p = a0p + (size_t)16 * lda;

  v8f acc00 = {0.f, 0.f, 0.f, 0.f, 0.f, 0.f, 0.f, 0.f};
  v8f acc01 = acc00, acc10 = acc00, acc11 = acc00;

  if (TRANS_B) {
    const float* b0p = B + (size_t)(tn + lid) * ldb + half * 2;
    const float* b1p = b0p + (size_t)16 * ldb;
    for (int k = 0; k < K; k += 32) {
      __builtin_prefetch(a0p + k + 64, 0, 3);   // global_prefetch_b8
      __builtin_prefetch(a1p + k + 64, 0, 3);
      __builtin_prefetch(b0p + k + 64, 0, 3);
      __builtin_prefetch(b1p + k + 64, 0, 3);
#pragma unroll
      for (int kk = 0; kk < 32; kk += 4) {
        const v2f a0 = *reinterpret_cast<const v2f*>(a0p + k + kk);
        const v2f a1 = *reinterpret_cast<const v2f*>(a1p + k + kk);
        const v2f b0 = *reinterpret_cast<const v2f*>(b0p + k + kk);
        const v2f b1 = *reinterpret_cast<const v2f*>(b1p + k + kk);
        acc00 = WMMA_F32(a0, b0, acc00);
        acc01 = WMMA_F32(a0, b1, acc01);
        acc10 = WMMA_F32(a1, b0, acc10);
        acc11 = WMMA_F32(a1, b1, acc11);
      }
    }
  } else {
    const float* bc = B + (size_t)(half * 2) * ldb + tn + lid;
    for (int k = 0; k < K; k += 32) {
      __builtin_prefetch(a0p + k + 64, 0, 3);
      __builtin_prefetch(a1p + k + 64, 0, 3);
#pragma unroll
      for (int kk = 0; kk < 32; kk += 4) {
        const v2f a0 = *reinterpret_cast<const v2f*>(a0p + k + kk);
        const v2f a1 = *reinterpret_cast<const v2f*>(a1p + k + kk);
        v2f b0, b1;
        const size_t r0 = (size_t)(k + kk) * ldb;
        const size_t r1 = (size_t)(k + kk + 1) * ldb;
        b0.x = bc[r0];
        b0.y = bc[r1];
        b1.x = bc[r0 + 16];
        b1.y = bc[r1 + 16];
        acc00 = WMMA_F32(a0, b0, acc00);
        acc01 = WMMA_F32(a0, b1, acc01);
        acc10 = WMMA_F32(a1, b0, acc10);
        acc11 = WMMA_F32(a1, b1, acc11);
      }
    }
  }

  const float bv0 = ADD_BIAS ? bias[tn + lid] : 0.f;
  const float bv1 = ADD_BIAS ? bias[tn + 16 + lid] : 0.f;
  store_tile16<ACCUM>(C, ldc, tm,      tn,      lid, half, acc00, alpha, bv0, beta);
  store_tile16<ACCUM>(C, ldc, tm,      tn + 16, lid, half, acc01, alpha, bv1, beta);
  store_tile16<ACCUM>(C, ldc, tm + 16, tn,      lid, half, acc10, alpha, bv0, beta);
  store_tile16<ACCUM>(C, ldc, tm + 16, tn + 16, lid, half, acc11, alpha, bv1, beta);
}

// ---------------------------------------------------------------------------
// Landmark means: q_l/k_l[h][m][d] = mean_{j<32} q/k[h][m*32+j][d] (unscaled;
// the 1/8 q-scale is folded into the logit GEMM alphas).
// grid = HEADS*MM_, block = 64 (d)
// ---------------------------------------------------------------------------
__global__ __launch_bounds__(64)
void landmark_means(const float* __restrict__ qkv,
                    float* __restrict__ q_l, float* __restrict__ k_l) {
  const int h = blockIdx.x >> 8;
  const int m = blockIdx.x & 255;
  const int d = threadIdx.x;
  const float* qb = qkv + (size_t)(m * LL_) * QKVLD_ + h * DHEAD_ + d;
  const float* kb = qb + 512;
  float sq = 0.f, sk = 0.f;
#pragma unroll 4
  for (int j = 0; j < LL_; ++j) {
    sq += qb[(size_t)j * QKVLD_];
    sk += kb[(size_t)j * QKVLD_];
  }
  const size_t o = ((size_t)h * MM_ + m) * DHEAD_ + d;
  q_l[o] = sq * (1.f / LL_);
  k_l[o] = sk * (1.f / LL_);
}

// ---------------------------------------------------------------------------
// In-place row softmax: grid = rows, block = 256
// ---------------------------------------------------------------------------
__global__ __launch_bounds__(256)
void softmax_rows(float* __restrict__ mat, int cols) {
  __shared__ float red[256];
  float* p = mat + (size_t)blockIdx.x * cols;
  const int t = threadIdx.x;
  float mx = -INFINITY;
  for (int c = t; c < cols; c += 256) mx = fmaxf(mx, p[c]);
  red[t] = mx;
  __syncthreads();
  for (int s = 128; s > 0; s >>= 1) {
    if (t < s) red[t] = fmaxf(red[t], red[t + s]);
    __syncthreads();
  }
  mx = red[0];
  __syncthreads();
  float sum = 0.f;
  for (int c = t; c < cols; c += 256) {
    float e = __expf(p[c] - mx);
    p[c] = e;
    sum += e;
  }
  red[t] = sum;
  __syncthreads();
  for (int s = 128; s > 0; s >>= 1) {
    if (t < s) red[t] += red[t + s];
    __syncthreads();
  }
  const float inv = 1.f / red[0];
  for (int c = t; c < cols; c += 256) p[c] *= inv;
}

// ---------------------------------------------------------------------------
// Pseudo-inverse init: z = x^T / (max_row_abs_sum * max_col_abs_sum)
// single block, 256 threads (x is 256x256)
// ---------------------------------------------------------------------------
__global__ __launch_bounds__(256)
void pinv_init(const float* __restrict__ x, float* __restrict__ z) {
  __shared__ float rr[256], cc[256];
  const int t = threadIdx.x;
  float rs = 0.f, cs = 0.f;
  for (int j = 0; j < 256; ++j) {
    rs += fabsf(x[t * 256 + j]);
    cs += fabsf(x[j * 256 + t]);
  }
  rr[t] = rs;
  cc[t] = cs;
  __syncthreads();
  for (int s = 128; s > 0; s >>= 1) {
    if (t < s) {
      rr[t] = fmaxf(rr[t], rr[t + s]);
      cc[t] = fmaxf(cc[t], cc[t + s]);
    }
    __syncthreads();
  }
  const float inv = 1.f / (rr[0] * cc[0]);
  for (int j = 0; j < 256; ++j) z[t * 256 + j] = x[j * 256 + t] * inv;
}

// out = c*I - in, 256x256 ; grid = 256, block = 256
__global__ __launch_bounds__(256)
void neg_plus_cI(float* __restrict__ out, const float* __restrict__ in, float c) {
  const int i = blockIdx.x, j = threadIdx.x;
  out[i * 256 + j] = (i == j ? c : 0.f) - in[i * 256 + j];
}

// ---------------------------------------------------------------------------
// Depthwise residual conv over sequence dim (K=33, pad 16), add into outh.
// outh layout [n, h*64+d]; v at qkv[n*1536 + 1024 + h*64 + d].
// one thread per (n, c=h*64+d)
// ---------------------------------------------------------------------------
__global__ __launch_bounds__(256)
void residual_conv_add(const float* __restrict__ qkv,
                       const float* __restrict__ kern,
                       float* __restrict__ outh) {
  const size_t idx = (size_t)blockIdx.x * 256 + threadIdx.x;
  const int n = (int)(idx >> 9);
  const int c = (int)(idx & 511);
  const int h = c >> 6;
  const float* v = qkv + 1024 + c;
  float s = 0.f;
#pragma unroll
  for (int kk = 0; kk < KCONV_; ++kk) {
    const int j = n + kk - (KCONV_ / 2);
    if (j >= 0 && j < NN_) s += kern[h * KCONV_ + kk] * v[(size_t)j * QKVLD_];
  }
  outh[idx] += s;
}

// ---------------------------------------------------------------------------
// Edge scores + segment sum: one wave per edge, 256-dim dot, atomic row add.
// ---------------------------------------------------------------------------
__global__ __launch_bounds__(256)
void edge_segment_sum(const float* __restrict__ q, const float* __restrict__ k,
                      const int* __restrict__ rows, const int* __restrict__ cols,
                      const float* __restrict__ vals, float* __restrict__ Araw,
                      int E) {
  const int wid = (int)((blockIdx.x * (size_t)blockDim.x + threadIdx.x) >> 5);
  const int lane = threadIdx.x & 31;
  if (wid >= E) return;
  const int r = rows[wid], c = cols[wid];
  const float* qr = q + (size_t)r * WP_;
  const float* kc = k + (size_t)c * WP_;
  float s = 0.f;
#pragma unroll
  for (int d = lane; d < WP_; d += 32) s += qr[d] * kc[d];
  for (int off = 16; off > 0; off >>= 1) s += __shfl_xor(s, off, 32);
  if (lane == 0) atomicAdd(Araw + r, s * 0.0625f * vals[wid]);  // WP^-0.5 = 1/16
}

__global__ __launch_bounds__(256)
void zero_vec(float* __restrict__ p, int n) {
  const int i = blockIdx.x * 256 + threadIdx.x;
  if (i < n) p[i] = 0.f;
}

__global__ __launch_bounds__(256)
void copy_vec(const float* __restrict__ src, float* __restrict__ dst, int n) {
  const int i = blockIdx.x * 256 + threadIdx.x;
  if (i < n) dst[i] = src[i];
}

// softmax over a single vector, single block
__global__ __launch_bounds__(1024)
void softmax_vec(const float* __restrict__ in, float* __restrict__ out, int n) {
  __shared__ float red[1024];
  const int t = threadIdx.x;
  float mx = -INFINITY;
  for (int i = t; i < n; i += 1024) mx = fmaxf(mx, in[i]);
  red[t] = mx;
  __syncthreads();
  for (int s = 512; s > 0; s >>= 1) {
    if (t < s) red[t] = fmaxf(red[t], red[t + s]);
    __syncthreads();
  }
  mx = red[0];
  __syncthreads();
  float sum = 0.f;
  for (int i = t; i < n; i += 1024) sum += __expf(in[i] - mx);
  red[t] = sum;
  __syncthreads();
  for (int s = 512; s > 0; s >>= 1) {
    if (t < s) red[t] += red[t + s];
    __syncthreads();
  }
  const float inv = 1.f / red[0];
  for (int i = t; i < n; i += 1024) out[i] = __expf(in[i] - mx) * inv;
}

// xl = alpha[i]*value ; wei = sigmoid(-xl) ; xo = xl*2*wei^2 + 2*enc*(1-wei^2)
__global__ __launch_bounds__(256)
void gating(const float* __restrict__ alpha, const float* __restrict__ value,
            const float* __restrict__ enc, float* __restrict__ xo) {
  const size_t idx = (size_t)blockIdx.x * 256 + threadIdx.x;
  const int i = (int)(idx >> 9);  // / D
  const float xl = alpha[i] * value[idx];
  const float wei = 1.f / (1.f + __expf(xl));  // sigmoid(-xl)
  const float sq = wei * wei;
  xo[idx] = xl * 2.f * sq + 2.f * enc[idx] * (1.f - sq);
}

// ---------------------------------------------------------------------------
// Host orchestration
// ---------------------------------------------------------------------------
extern "C" void kernel_launch(void* const* d_in, const int* in_sizes, int n_in,
                              void* d_out, int out_size, void* d_ws, size_t ws_size,
                              hipStream_t stream) {
  const float* dense   = (const float*)d_in[0];
  const int*   adjr    = (const int*)d_in[1];
  const int*   adjc    = (const int*)d_in[2];
  const float* adjv    = (const float*)d_in[3];
  const float* wq      = (const float*)d_in[4];
  const float* wk      = (const float*)d_in[5];
  const float* w_qkv   = (const float*)d_in[6];
  const float* w_out   = (const float*)d_in[7];
  const float* b_out   = (const float*)d_in[8];
  const float* res_k   = (const float*)d_in[9];
  const float* wv_w    = (const float*)d_in[10];
  const float* wv_b    = (const float*)d_in[11];
  const int E = in_sizes[1];

  float* xo_out = (float*)d_out;                 // [N, D]
  float* Araw_out = xo_out + (size_t)NN_ * DD_;  // [N]

  // workspace carve-up
  float* w = (float*)d_ws;
  float* qkv  = w; w += (size_t)NN_ * QKVLD_;
  float* q_l  = w; w += (size_t)HEADS_ * MM_ * DHEAD_;
  float* k_l  = w; w += (size_t)HEADS_ * MM_ * DHEAD_;
  float* a1   = w; w += (size_t)NN_ * MM_;       // per-head, reused
  float* a2   = w; w += (size_t)MM_ * MM_;
  float* a3   = w; w += (size_t)MM_ * NN_;
  float* zb   = w; w += (size_t)MM_ * MM_;
  float* znb  = w; w += (size_t)MM_ * MM_;
  float* xz   = w; w += (size_t)MM_ * MM_;
  float* tb   = w; w += (size_t)MM_ * MM_;
  float* tmp2 = w; w += (size_t)MM_ * MM_;
  float* a3v  = w; w += (size_t)MM_ * DHEAD_;
  float* a1z  = w; w += (size_t)NN_ * MM_;
  float* outh = w; w += (size_t)NN_ * DD_;
  float* enc  = w; w += (size_t)NN_ * DD_;
  float* qe   = w; w += (size_t)NN_ * WP_;
  float* ke   = w; w += (size_t)NN_ * WP_;
  float* Araw = w; w += NN_;
  float* alph = w; w += NN_;
  float* valb = w; w += (size_t)NN_ * DD_;
  (void)ws_size; (void)n_in; (void)out_size;

  const dim3 blk32(32);

  // 1) qkv = dense @ w_qkv^T      [8192x512] x [1536x512]^T
  wmma_gemm_f32<1, 0, 0><<<dim3(QKVLD_ / 32, NN_ / 32), blk32, 0, stream>>>(
      dense, DD_, w_qkv, DD_, qkv, QKVLD_, nullptr, DD_, 1.f, 0.f);

  // 2) landmark means
  landmark_means<<<dim3(HEADS_ * MM_), dim3(64), 0, stream>>>(qkv, q_l, k_l);

  // 3) per-head Nystrom
  for (int h = 0; h < HEADS_; ++h) {
    const float* qh  = qkv + h * DHEAD_;           // [N,64]  lda 1536
    const float* kh  = qkv + 512 + h * DHEAD_;     // [N,64]  ldb 1536
    const float* vh  = qkv + 1024 + h * DHEAD_;    // [N,64]  ldb 1536
    const float* qlh = q_l + (size_t)h * MM_ * DHEAD_;  // [256,64]
    const float* klh = k_l + (size_t)h * MM_ * DHEAD_;

    // a1 = softmax( 0.125 * q @ k_l^T )   [N,256]
    wmma_gemm_f32<1, 0, 0><<<dim3(MM_ / 32, NN_ / 32), blk32, 0, stream>>>(
        qh, QKVLD_, klh, DHEAD_, a1, MM_, nullptr, DHEAD_, 0.125f, 0.f);
    softmax_rows<<<dim3(NN_), dim3(256), 0, stream>>>(a1, MM_);

    // a2 = softmax( 0.125 * q_l @ k_l^T )  [256,256]
    wmma_gemm_f32<1, 0, 0><<<dim3(MM_ / 32, MM_ / 32), blk32, 0, stream>>>(
        qlh, DHEAD_, klh, DHEAD_, a2, MM_, nullptr, DHEAD_, 0.125f, 0.f);
    softmax_rows<<<dim3(MM_), dim3(256), 0, stream>>>(a2, MM_);

    // a3 = softmax( 0.125 * q_l @ k^T )    [256,N]
    wmma_gemm_f32<1, 0, 0><<<dim3(NN_ / 32, MM_ / 32), blk32, 0, stream>>>(
        qlh, DHEAD_, kh, QKVLD_, a3, NN_, nullptr, DHEAD_, 0.125f, 0.f);
    softmax_rows<<<dim3(MM_), dim3(256), 0, stream>>>(a3, NN_);

    // pinv(a2), 6 Newton-Schulz-style iterations
    pinv_init<<<dim3(1), dim3(256), 0, stream>>>(a2, zb);
    float* z = zb;
    float* zn = znb;
    for (int it = 0; it < 6; ++it) {
      wmma_gemm_f32<0, 0, 0><<<dim3(8, 8), blk32, 0, stream>>>(
          a2, MM_, z, MM_, xz, MM_, nullptr, MM_, 1.f, 0.f);
      neg_plus_cI<<<dim3(256), dim3(256), 0, stream>>>(tb, xz, 7.f);
      wmma_gemm_f32<0, 0, 0><<<dim3(8, 8), blk32, 0, stream>>>(
          xz, MM_, tb, MM_, tmp2, MM_, nullptr, MM_, 1.f, 0.f);
      neg_plus_cI<<<dim3(256), dim3(256), 0, stream>>>(tb, tmp2, 15.f);
      wmma_gemm_f32<0, 0, 0><<<dim3(8, 8), blk32, 0, stream>>>(
          xz, MM_, tb, MM_, tmp2, MM_, nullptr, MM_, 1.f, 0.f);
      neg_plus_cI<<<dim3(256), dim3(256), 0, stream>>>(tb, tmp2, 13.f);
      wmma_gemm_f32<0, 0, 0><<<dim3(8, 8), blk32, 0, stream>>>(
          z, MM_, tb, MM_, zn, MM_, nullptr, MM_, 0.25f, 0.f);
      float* t = z; z = zn; zn = t;
    }

    // a1z = a1 @ z                [N,256]
    wmma_gemm_f32<0, 0, 0><<<dim3(MM_ / 32, NN_ / 32), blk32, 0, stream>>>(
        a1, MM_, z, MM_, a1z, MM_, nullptr, MM_, 1.f, 0.f);
    // a3v = a3 @ v                [256,64]
    wmma_gemm_f32<0, 0, 0><<<dim3(DHEAD_ / 32, MM_ / 32), blk32, 0, stream>>>(
        a3, NN_, vh, QKVLD_, a3v, DHEAD_, nullptr, NN_, 1.f, 0.f);
    // outh[:, h*64:(h+1)*64] = a1z @ a3v
    wmma_gemm_f32<0, 0, 0><<<dim3(DHEAD_ / 32, NN_ / 32), blk32, 0, stream>>>(
        a1z, MM_, a3v, DHEAD_, outh + h * DHEAD_, DD_, nullptr, MM_, 1.f, 0.f);
  }

  // 4) residual depthwise conv over sequence, added into outh
  residual_conv_add<<<dim3((NN_ * DD_) / 256), dim3(256), 0, stream>>>(
      qkv, res_k, outh);

  // 5) enc = outh @ w_out^T + b_out + dense   (seed enc with dense, accumulate)
  hipMemcpyAsync(enc, dense, (size_t)NN_ * DD_ * sizeof(float),
                 hipMemcpyDeviceToDevice, stream);
  wmma_gemm_f32<1, 1, 1><<<dim3(DD_ / 32, NN_ / 32), blk32, 0, stream>>>(
      outh, DD_, w_out, DD_, enc, DD_, b_out, DD_, 1.f, 1.f);

  // 6) q = enc @ wq, k = enc @ wk   (NN: B is [512,256] row-major)
  wmma_gemm_f32<0, 0, 0><<<dim3(WP_ / 32, NN_ / 32), blk32, 0, stream>>>(
      enc, DD_, wq, WP_, qe, WP_, nullptr, DD_, 1.f, 0.f);
  wmma_gemm_f32<0, 0, 0><<<dim3(WP_ / 32, NN_ / 32), blk32, 0, stream>>>(
      enc, DD_, wk, WP_, ke, WP_, nullptr, DD_, 1.f, 0.f);

  // 7) edge scores -> segment row sums
  zero_vec<<<dim3(NN_ / 256), dim3(256), 0, stream>>>(Araw, NN_);
  edge_segment_sum<<<dim3((E * 32 + 255) / 256), dim3(256), 0, stream>>>(
      qe, ke, adjr, adjc, adjv, Araw, E);

  // 8) alpha = softmax over all N
  softmax_vec<<<dim3(1), dim3(1024), 0, stream>>>(Araw, alph, NN_);

  // 9) value = dense @ wv_w^T + wv_b
  wmma_gemm_f32<1, 1, 0><<<dim3(DD_ / 32, NN_ / 32), blk32, 0, stream>>>(
      dense, DD_, wv_w, DD_, valb, DD_, wv_b, DD_, 1.f, 0.f);

  // 10) gated blend -> xo ; append A_raw
  gating<<<dim3((NN_ * DD_) / 256), dim3(256), 0, stream>>>(
      alph, valb, enc, xo_out);
  copy_vec<<<dim3(NN_ / 256), dim3(256), 0, stream>>>(Araw, Araw_out, NN_);
}